// DetecLoss_90503550861535
// MI455X (gfx1250) — compile-verified
//
#include <hip/hip_runtime.h>
#include <hip/hip_bf16.h>
#include <stdint.h>

typedef float v2f  __attribute__((ext_vector_type(2)));
typedef float v8f  __attribute__((ext_vector_type(8)));
typedef unsigned int u32x4 __attribute__((ext_vector_type(4)));
typedef int   i32x8 __attribute__((ext_vector_type(8)));
typedef int   i32x4 __attribute__((ext_vector_type(4)));

#define THRESH    0.5f
#define NEG_RATIO 3
#define VAR0      0.1f
#define VAR1      0.2f

// ---- order-preserving float->uint map (monotonic for all floats) ----
__device__ __forceinline__ uint32_t fkey(float f) {
  uint32_t u = __float_as_uint(f);
  return (u & 0x80000000u) ? ~u : (u | 0x80000000u);
}

// ---- exact f32 reduction of 64 values (2/lane) via two chained WMMAs ----
// Stage 1: A[m,k] = data (K0=x lanes0-15, K1=y lanes0-15, K2=x lanes16-31,
//          K3=y lanes16-31), B = ones  ->  D[m,n] = rowsum R[m] (bcast over n).
//          t = sum(c[0..7]):  lanes 0-15 = sum R[0..7], lanes 16-31 = sum R[8..15].
// Stage 2: A2 = (t,0), B = ones -> c2[0] = t[m] + t[m+16] = grand total, all lanes.
// Exact (x*1.0 products, f32 RNE adds), deterministic, zero DS/shuffle traffic.
__device__ __forceinline__ float wave_sum64(float x, float y) {
  v2f ones; ones.x = 1.0f; ones.y = 1.0f;
  v2f a;    a.x = x;       a.y = y;
  v8f c = {};
  c = __builtin_amdgcn_wmma_f32_16x16x4_f32(false, a, false, ones, (short)0, c, false, false);
  float t = ((c[0] + c[1]) + (c[2] + c[3])) + ((c[4] + c[5]) + (c[6] + c[7]));
  v2f a2;   a2.x = t;      a2.y = 0.0f;
  v8f c2 = {};
  c2 = __builtin_amdgcn_wmma_f32_16x16x4_f32(false, a2, false, ones, (short)0, c2, false, false);
  return c2[0];   // total of all 64 inputs, broadcast to every lane
}

template <int NW>
__device__ __forceinline__ float block_sum_f32_pair(float x, float y, float* sbuf, int tid) {
  float w = wave_sum64(x, y);
  if ((tid & 31) == 0) sbuf[tid >> 5] = w;
  __syncthreads();
  float r = 0.0f;
  if (tid == 0) { for (int i = 0; i < NW; ++i) r += sbuf[i]; }
  __syncthreads();
  return r;  // valid on tid 0
}

template <int NW>
__device__ __forceinline__ int block_sum_i32_bcast(int v, float* sbuf, int tid) {
  // counts <= 32 per lane, block totals < 2^24: exact in f32
  float w = wave_sum64((float)v, 0.0f);
  if ((tid & 31) == 0) sbuf[tid >> 5] = w;
  __syncthreads();
  if (tid == 0) { float r = 0.0f; for (int i = 0; i < NW; ++i) r += sbuf[i]; sbuf[0] = r; }
  __syncthreads();
  int r = (int)sbuf[0];
  __syncthreads();
  return r;  // broadcast to all threads
}

// ---- async stage of one image's targets (n floats) into LDS --------------
// wave 0 issues per-lane global_load_async_to_lds_b32 (GVS form), waits on
// ASYNCcnt, then the whole block syncs. (accepted by gfx1250 assembler)
__device__ __forceinline__ void stage_targets_async(const float* targets_row,
                                                    float* straw, int n, int tid) {
  if (tid < 32) {
    uint64_t base = (uint64_t)(uintptr_t)targets_row;
    for (int j = tid; j < n; j += 32) {
      uint32_t ldsa = (uint32_t)(uintptr_t)&straw[j];   // low 32 bits = LDS offset
      uint32_t goff = (uint32_t)(j * 4);
      asm volatile("global_load_async_to_lds_b32 %0, %1, %2"
                   :: "v"(ldsa), "v"(goff), "s"(base) : "memory");
    }
    asm volatile("s_wait_asynccnt 0" ::: "memory");
  }
  __syncthreads();
}

// ---- TDM stage: DMA one image's targets (n floats) into LDS --------------
// Tensor DMA Descriptor per CDNA5 ISA:
//   group0: count=1 | lds_addr | global_addr[56:0] | type=2
//   group1: data_size=4B, tensor_dim0=n, tensor_dim1=1, tile_dim0=n,
//           tile_dim1=1, tensor_dim0_stride=n; groups 2/3 zero (<=2D tensor).
// Wave-scoped (EXEC ignored): issued + waited by wave 0, then block barrier.
__device__ __forceinline__ void stage_targets_tdm(const float* targets_row,
                                                  float* straw, int n, int tid) {
#if __has_builtin(__builtin_amdgcn_tensor_load_to_lds)
  if (tid < 32) {
    uint64_t ga = (uint64_t)(uintptr_t)targets_row;
    uint32_t la = (uint32_t)(uintptr_t)straw;           // LDS byte address
    u32x4 g0;
    g0.x = 1u;                                          // count=1, user mode
    g0.y = la;                                          // lds_addr
    g0.z = (uint32_t)ga;                                // global_addr[31:0]
    g0.w = (uint32_t)((ga >> 32) & 0x01FFFFFFu) | (2u << 30);  // addr[56:32], type=2
    i32x8 g1;
    g1[0] = 0x00020000;                                 // data_size=2 (4 bytes)
    g1[1] = (n & 0xFFFF) << 16;                         // tensor_dim0[15:0]
    g1[2] = 1 << 16;                                    // tensor_dim1 = 1
    g1[3] = (n & 0xFFFF) << 16;                         // tile_dim0 = n
    g1[4] = 1;                                          // tile_dim1 = 1
    g1[5] = n;                                          // tensor_dim0_stride
    g1[6] = 0;
    g1[7] = 0;
    i32x4 gz4 = {0, 0, 0, 0};
    i32x8 gz8 = {0, 0, 0, 0, 0, 0, 0, 0};
    __builtin_amdgcn_tensor_load_to_lds(g0, g1, gz4, gz4, gz8, 0);  // 6-arg form
    __builtin_amdgcn_s_wait_tensorcnt(0);
  }
  __syncthreads();
#else
  stage_targets_async(targets_row, straw, n, tid);
#endif
}

// ------------------- kernel 0: zero scratch accumulators -------------------
__global__ void k_init(unsigned long long* btruth, int* num_pos, float* acc,
                       int nBT, int B) {
  int i = blockIdx.x * blockDim.x + threadIdx.x;
  if (i < nBT) btruth[i] = 0ull;
  if (i < B)   num_pos[i] = 0;
  if (i < 2)   acc[i] = 0.0f;
}

// ------------- kernel 1: priors -> point form + area (once, reused B×) -----
__global__ void k_prep(const float* __restrict__ priors, float* __restrict__ pt,
                       float* __restrict__ area_p, int P) {
  int p = blockIdx.x * blockDim.x + threadIdx.x;
  if (p >= P) return;
  float cx = priors[p * 4 + 0], cy = priors[p * 4 + 1];
  float w  = priors[p * 4 + 2], h  = priors[p * 4 + 3];
  float x0 = cx - w * 0.5f, y0 = cy - h * 0.5f;
  float x1 = cx + w * 0.5f, y1 = cy + h * 0.5f;
  pt[p * 4 + 0] = x0; pt[p * 4 + 1] = y0;
  pt[p * 4 + 2] = x1; pt[p * 4 + 3] = y1;
  area_p[p] = (x1 - x0) * (y1 - y0);  // same expression as reference
}

// --------- kernel 2: IoU, per-prior best truth, per-truth best prior -------
__global__ void k_match(const float* __restrict__ targets, const float* __restrict__ pt,
                        const float* __restrict__ area_p, float* __restrict__ bto,
                        int* __restrict__ bti, unsigned long long* __restrict__ btruth,
                        int P, int M) {
  __shared__ float straw[32 * 5];                 // raw targets for this image
  __shared__ unsigned long long skey[32];         // per-truth (ov,~p) max
  int tid = threadIdx.x;
  int b   = blockIdx.y;
  int p   = blockIdx.x * blockDim.x + tid;

  if (tid < 32) skey[tid] = 0ull;
  stage_targets_tdm(targets + (size_t)b * M * 5, straw, M * 5, tid);  // TDM path

  // stream-ahead: prefetch next prior tile (speculative, OOB-safe)
  __builtin_prefetch(pt + (size_t)(p + blockDim.x) * 4, 0, 1);

  float px0 = pt[p * 4 + 0], py0 = pt[p * 4 + 1];
  float px1 = pt[p * 4 + 2], py1 = pt[p * 4 + 3];
  float ap  = area_p[p];

  float best = -2.0f;
  int   bestt = 0;

  for (int t = 0; t < M; ++t) {
    float b0 = straw[t * 5 + 0], b1 = straw[t * 5 + 1];
    float b2 = straw[t * 5 + 2], b3 = straw[t * 5 + 3];
    float lx = fmaxf(b0, px0);
    float ly = fmaxf(b1, py0);
    float rx = fminf(b2, px1);
    float ry = fminf(b3, py1);
    float iw = fmaxf(rx - lx, 0.0f), ih = fmaxf(ry - ly, 0.0f);
    float inter = iw * ih;
    float at = (b2 - b0) * (b3 - b1);
    float ov = inter / (at + ap - inter);
    ov = ((b2 - b0) > 0.0f) ? ov : -1.0f;
    if (ov > best) { best = ov; bestt = t; }  // first-index tie-break (argmax axis=0)

    // per-truth argmax over p: non-returning LDS atomic, key = (ordered ov, ~p)
    unsigned long long key =
        ((unsigned long long)fkey(ov) << 32) | (uint32_t)(~(uint32_t)p);
    atomicMax(&skey[t], key);
  }

  bto[(size_t)b * P + p] = best;
  bti[(size_t)b * P + p] = bestt;

  __syncthreads();
  if (tid < M) atomicMax(&btruth[b * M + tid], skey[tid]);
}

// ------------- kernel 3: force-match override (sequential t, last wins) ----
__global__ void k_force(const float* __restrict__ targets,
                        const unsigned long long* __restrict__ btruth,
                        float* __restrict__ bto, int* __restrict__ bti, int P, int M) {
  int b = blockIdx.x;
  if (threadIdx.x != 0) return;
  for (int t = 0; t < M; ++t) {
    const float* tp = targets + (size_t)(b * M + t) * 5;
    if (!((tp[2] - tp[0]) > 0.0f)) continue;
    uint32_t p = ~(uint32_t)btruth[b * M + t];
    bto[(size_t)b * P + p] = 2.0f;
    bti[(size_t)b * P + p] = t;
  }
}

// ------------- kernel 4: smooth-L1 loc loss + positive counts --------------
__global__ void k_locloss(const float* __restrict__ loc, const float* __restrict__ priors,
                          const float* __restrict__ targets, const float* __restrict__ bto,
                          const int* __restrict__ bti, int* __restrict__ num_pos,
                          float* __restrict__ acc, int P, int M) {
  __shared__ float straw[32 * 5];
  __shared__ float sfbuf[8];
  int tid = threadIdx.x;
  int b   = blockIdx.y;
  int p   = blockIdx.x * blockDim.x + tid;

  stage_targets_async(targets + (size_t)b * M * 5, straw, M * 5, tid);  // async path

  float v  = bto[(size_t)b * P + p];
  bool pos = v >= THRESH;
  int  t   = bti[(size_t)b * P + p];

  float m0 = straw[t * 5 + 0], m1 = straw[t * 5 + 1];
  float m2 = straw[t * 5 + 2], m3 = straw[t * 5 + 3];
  float cx = priors[p * 4 + 0], cy = priors[p * 4 + 1];
  float pw = priors[p * 4 + 2], ph = priors[p * 4 + 3];

  float g0 = ((m0 + m2) * 0.5f - cx) / (VAR0 * pw);
  float g1 = ((m1 + m3) * 0.5f - cy) / (VAR0 * ph);
  float g2 = logf(fmaxf((m2 - m0) / pw, 1e-8f)) / VAR1;
  float g3 = logf(fmaxf((m3 - m1) / ph, 1e-8f)) / VAR1;

  const float* l = loc + ((size_t)b * P + p) * 4;
  float sa = 0.0f, sb = 0.0f, d;
  d = fabsf(l[0] - g0); sa += (d < 1.0f) ? 0.5f * d * d : d - 0.5f;
  d = fabsf(l[1] - g1); sa += (d < 1.0f) ? 0.5f * d * d : d - 0.5f;
  d = fabsf(l[2] - g2); sb += (d < 1.0f) ? 0.5f * d * d : d - 0.5f;
  d = fabsf(l[3] - g3); sb += (d < 1.0f) ? 0.5f * d * d : d - 0.5f;
  if (!pos) { sa = 0.0f; sb = 0.0f; }

  float bs = block_sum_f32_pair<8>(sa, sb, sfbuf, tid);       // WMMA reduce
  int   pc = block_sum_i32_bcast<8>(pos ? 1 : 0, sfbuf, tid); // WMMA reduce
  if (tid == 0) {
    atomicAdd(&acc[0], bs);
    atomicAdd(&num_pos[b], pc);
  }
}

// ---- kernel 5: exact stable hard-negative mining + conf loss (1 blk/img) --
#define K5_THREADS 1024
#define MAXCHUNK   32

__global__ __launch_bounds__(K5_THREADS) void k_neg_conf(
    const float* __restrict__ conf, const float* __restrict__ bto,
    const int* __restrict__ num_pos, float* __restrict__ acc, int P) {
  __shared__ int   sscan[K5_THREADS];
  __shared__ float sfbuf[32];
  int tid = threadIdx.x;
  int b   = blockIdx.x;
  int chunk = P / K5_THREADS;  // 32 for P=32768
  int base  = tid * chunk;

  const float* row = bto + (size_t)b * P;
  uint32_t keys[MAXCHUNK];
  uint32_t candmask = 0;
  for (int i = 0; i < chunk; ++i) {
    float v = row[base + i];
    if (!(v >= THRESH)) candmask |= (1u << i);   // candidate = not positive
    keys[i] = fkey(v);                           // monotonic key, ascending bto
  }

  int npos  = num_pos[b];
  int ncand = P - npos;
  int k = NEG_RATIO * npos;
  if (k > ncand) k = ncand;

  uint32_t T = 0u;
  int need_eq = 0;
  if (k > 0) {
    // bitwise radix-select of the k-th smallest candidate key
    uint32_t prefix = 0;
    uint32_t want = (uint32_t)k;
    for (int bit = 31; bit >= 0; --bit) {
      int cnt = 0;
      for (int i = 0; i < chunk; ++i) {
        if (!((candmask >> i) & 1u)) continue;
        uint32_t x = keys[i] ^ prefix;
        if ((((unsigned long long)x) >> (bit + 1)) == 0ull &&
            (((keys[i] >> bit) & 1u) == 0u)) cnt++;
      }
      int total = block_sum_i32_bcast<32>(cnt, sfbuf, tid);   // WMMA reduce
      if ((int)want > total) { want -= (uint32_t)total; prefix |= (1u << bit); }
    }
    T = prefix;
    int lcnt = 0;
    for (int i = 0; i < chunk; ++i)
      if (((candmask >> i) & 1u) && keys[i] < T) lcnt++;
    int n_less = block_sum_i32_bcast<32>(lcnt, sfbuf, tid);
    need_eq = k - n_less;  // >= 1 by construction
  }

  // stable tie ranking: exclusive scan of per-thread equal-key counts
  int leq = 0;
  if (k > 0)
    for (int i = 0; i < chunk; ++i)
      if (((candmask >> i) & 1u) && keys[i] == T) leq++;
  sscan[tid] = leq;
  __syncthreads();
  for (int off = 1; off < K5_THREADS; off <<= 1) {
    int tmp = (tid >= off) ? sscan[tid - off] : 0;
    __syncthreads();
    sscan[tid] += tmp;
    __syncthreads();
  }
  int running = sscan[tid] - leq;

  // conf loss over pos | neg (two accumulators feed the pair-WMMA reduce)
  float a0 = 0.0f, a1 = 0.0f;
  for (int i = 0; i < chunk; ++i) {
    int p = base + i;
    bool cand = ((candmask >> i) & 1u) != 0;
    bool pos  = !cand;
    bool neg  = false;
    if (k > 0 && cand) {
      if (keys[i] < T) neg = true;
      else if (keys[i] == T) { if (running < need_eq) neg = true; running++; }
    }
    if (pos || neg) {
      const float* cp = conf + ((size_t)b * P + p) * 2;
      float x0 = cp[0], x1 = cp[1];
      float mm = fmaxf(x0, x1);
      float lse = mm + logf(expf(x0 - mm) + expf(x1 - mm));
      float picked = (pos ? x1 : x0) - lse;   // picked log-softmax
      if (i & 1) a1 += picked; else a0 += picked;
    }
  }
  float tot = block_sum_f32_pair<32>(a0, a1, sfbuf, tid);  // WMMA reduce
  if (tid == 0) atomicAdd(&acc[1], tot);
}

// ------------------------------ kernel 6: finalize -------------------------
__global__ void k_final(const int* __restrict__ num_pos, const float* __restrict__ acc,
                        float* __restrict__ out, int B) {
  if (blockIdx.x == 0 && threadIdx.x == 0) {
    int N = 0;
    for (int i = 0; i < B; ++i) N += num_pos[i];
    float Nf = (float)N;
    out[0] = acc[0] / Nf;
    out[1] = -acc[1] / Nf;
  }
}

extern "C" void kernel_launch(void* const* d_in, const int* in_sizes, int n_in,
                              void* d_out, int out_size, void* d_ws, size_t ws_size,
                              hipStream_t stream) {
  const float* loc     = (const float*)d_in[0];
  const float* conf    = (const float*)d_in[1];
  const float* priors  = (const float*)d_in[2];
  const float* targets = (const float*)d_in[3];

  int P = in_sizes[2] / 4;            // 32768
  int B = in_sizes[0] / (P * 4);      // 32
  int M = in_sizes[3] / (B * 5);      // 32

  char* ws = (char*)d_ws;
  size_t off = 0;
  float* pt     = (float*)(ws + off); off += (size_t)P * 4 * sizeof(float);
  float* areap  = (float*)(ws + off); off += (size_t)P * sizeof(float);
  float* bto    = (float*)(ws + off); off += (size_t)B * P * sizeof(float);
  int*   bti    = (int*)(ws + off);   off += (size_t)B * P * sizeof(int);
  off = (off + 7) & ~(size_t)7;
  unsigned long long* btruth = (unsigned long long*)(ws + off); off += (size_t)B * M * 8;
  int*   num_pos = (int*)(ws + off);  off += (size_t)B * sizeof(int);
  float* acc     = (float*)(ws + off); off += 2 * sizeof(float);

  k_init    <<<dim3((B * M + 255) / 256), 256, 0, stream>>>(btruth, num_pos, acc, B * M, B);
  k_prep    <<<dim3((P + 255) / 256),     256, 0, stream>>>(priors, pt, areap, P);
  k_match   <<<dim3(P / 256, B),          256, 0, stream>>>(targets, pt, areap, bto, bti, btruth, P, M);
  k_force   <<<dim3(B),                    32, 0, stream>>>(targets, btruth, bto, bti, P, M);
  k_locloss <<<dim3(P / 256, B),          256, 0, stream>>>(loc, priors, targets, bto, bti, num_pos, acc, P, M);
  k_neg_conf<<<dim3(B),            K5_THREADS, 0, stream>>>(conf, bto, num_pos, acc, P);
  k_final   <<<1, 32, 0, stream>>>(num_pos, acc, (float*)d_out, B);
}